// GConv_23038204576432
// MI455X (gfx1250) — compile-verified
//
#include <hip/hip_runtime.h>

typedef __attribute__((ext_vector_type(2))) float v2f;
typedef __attribute__((ext_vector_type(4))) float v4f;
typedef __attribute__((ext_vector_type(8))) float v8f;

#define FDIM 128   // feature dim for both layers (reference: D = H = 128)

// ---------------- degree / normalization ----------------

__global__ void k_deg_init(float* __restrict__ deg, int N) {
  int i = blockIdx.x * blockDim.x + threadIdx.x;
  if (i < N) deg[i] = 1.0f;              // self loop contributes 1
}

__global__ void k_deg_scatter(const int* __restrict__ dst, float* __restrict__ deg, int E) {
  int e = blockIdx.x * blockDim.x + threadIdx.x;
  if (e < E) atomicAdd(&deg[dst[e]], 1.0f);
}

__global__ void k_deg_finalize(float* __restrict__ deg, int N) {
  int i = blockIdx.x * blockDim.x + threadIdx.x;
  if (i < N) deg[i] = rsqrtf(deg[i]);    // deg >= 1 always (self loops)
}

// ---------------- dense GEMM via fp32 WMMA ----------------
// out[N x 128] = A[N x 128] @ W[128 x 128]
// One 16x16 output tile per wave; 8 waves/block cover all 128 output columns
// for one 16-row stripe. K loop in steps of 4 -> v_wmma_f32_16x16x4_f32.

__global__ __launch_bounds__(256)
void k_gemm_wmma(const float* __restrict__ A, const float* __restrict__ W,
                 float* __restrict__ out, int N) {
  const int lane   = threadIdx.x & 31;
  const int wave   = threadIdx.x >> 5;
  const int m_base = blockIdx.x * 16;
  const int n_base = wave * 16;

  const int l15   = lane & 15;          // M for A-frag, N for B-frag
  const int khalf = (lane >> 4) * 2;    // lanes 0-15: K={0,1}; lanes 16-31: K={2,3}

  int m = m_base + l15;
  if (m >= N) m = N - 1;                // clamp loads; store is guarded
  const float* arow = A + (size_t)m * FDIM;
  const float* wcol = W + (n_base + l15);

  v8f acc = {};
  #pragma unroll
  for (int k = 0; k < FDIM; k += 4) {
    v2f a, b;
    a.x = arow[k + khalf];
    a.y = arow[k + khalf + 1];
    b.x = wcol[(size_t)(k + khalf) * FDIM];
    b.y = wcol[(size_t)(k + khalf + 1) * FDIM];
    acc = __builtin_amdgcn_wmma_f32_16x16x4_f32(
        /*neg_a=*/false, a, /*neg_b=*/false, b,
        /*c_mod=*/(short)0, acc, /*reuse_a=*/false, /*reuse_b=*/false);
  }

  // C/D layout: VGPR v, lanes 0-15 -> M=v; lanes 16-31 -> M=8+v; N = lane&15
  const int row0 = (lane >> 4) * 8;
  const int col  = n_base + l15;
  #pragma unroll
  for (int v = 0; v < 8; ++v) {
    int r = m_base + row0 + v;
    if (r < N) out[(size_t)r * FDIM + col] = acc[v];
  }
}

// ---------------- aggregation ----------------

// agg[i][:] = h[i][:] * dis[i]^2   (zero-init fused with self-loop term)
__global__ void k_agg_init(const float* __restrict__ h, const float* __restrict__ dis,
                           float* __restrict__ agg, int total) {
  int idx = blockIdx.x * blockDim.x + threadIdx.x;
  if (idx < total) {
    float d = dis[idx >> 7];            // FDIM == 128
    agg[idx] = h[idx] * (d * d);
  }
}

// one wave (32 lanes) per edge; lane j: float4 gather of h[src][4j..4j+3],
// scale by dis[src]*dis[dst], 4x global_atomic_add_f32 into agg[dst]
__global__ __launch_bounds__(256)
void k_edge_scatter(const int* __restrict__ src, const int* __restrict__ dst,
                    const float* __restrict__ dis,
                    const float* __restrict__ h, float* __restrict__ agg, int E) {
  unsigned gid = blockIdx.x * blockDim.x + threadIdx.x;
  int e    = (int)(gid >> 5);
  int lane = (int)(gid & 31);
  if (e >= E) return;
  int s = src[e];
  int d = dst[e];
  float norm = dis[s] * dis[d];
  const v4f* hv = (const v4f*)(h + (size_t)s * FDIM);
  v4f val = hv[lane];
  float* ad = agg + (size_t)d * FDIM + lane * 4;
  atomicAdd(ad + 0, val.x * norm);
  atomicAdd(ad + 1, val.y * norm);
  atomicAdd(ad + 2, val.z * norm);
  atomicAdd(ad + 3, val.w * norm);
}

// out = relu(agg + bias)
__global__ void k_bias_relu(const float* __restrict__ agg, const float* __restrict__ bias,
                            float* __restrict__ out, int total) {
  int idx = blockIdx.x * blockDim.x + threadIdx.x;
  if (idx < total) {
    float v = agg[idx] + bias[idx & (FDIM - 1)];
    out[idx] = v > 0.0f ? v : 0.0f;
  }
}

// ---------------- host orchestration ----------------

extern "C" void kernel_launch(void* const* d_in, const int* in_sizes, int n_in,
                              void* d_out, int out_size, void* d_ws, size_t ws_size,
                              hipStream_t stream) {
  const float* x  = (const float*)d_in[0];
  const int*   ei = (const int*)d_in[1];    // edge_index [2, E]
  const float* W1 = (const float*)d_in[2];
  const float* b1 = (const float*)d_in[3];
  const float* W2 = (const float*)d_in[4];
  const float* b2 = (const float*)d_in[5];

  const int N = in_sizes[0] / FDIM;
  const int E = in_sizes[1] / 2;
  const int* src = ei;
  const int* dst = ei + E;

  char*  ws = (char*)d_ws;
  size_t nf = (size_t)N * FDIM * sizeof(float);
  float* bufA = (float*)ws;                 // GEMM out / ping
  float* bufB = (float*)(ws + nf);          // agg / pong
  float* dis  = (float*)(ws + 2 * nf);      // deg -> deg^{-1/2}

  const int t = 256;
  const int total   = N * FDIM;
  const unsigned gN = (N + t - 1) / t;
  const unsigned gE = (E + t - 1) / t;
  const unsigned gT = (total + t - 1) / t;
  const unsigned gS = (unsigned)(((size_t)E * 32 + t - 1) / t);
  const unsigned mtiles = (N + 15) / 16;

  // normalization coefficients
  k_deg_init    <<<gN, t, 0, stream>>>(dis, N);
  k_deg_scatter <<<gE, t, 0, stream>>>(dst, dis, E);
  k_deg_finalize<<<gN, t, 0, stream>>>(dis, N);

  // layer 1: x @ W1 -> bufA ; aggregate -> bufB ; relu -> bufA (z1)
  k_gemm_wmma   <<<mtiles, 256, 0, stream>>>(x, W1, bufA, N);
  k_agg_init    <<<gT, t, 0, stream>>>(bufA, dis, bufB, total);
  k_edge_scatter<<<gS, t, 0, stream>>>(src, dst, dis, bufA, bufB, E);
  k_bias_relu   <<<gT, t, 0, stream>>>(bufB, b1, bufA, total);

  // layer 2: z1 @ W2 -> bufB ; aggregate -> bufA ; relu -> d_out
  k_gemm_wmma   <<<mtiles, 256, 0, stream>>>(bufA, W2, bufB, N);
  k_agg_init    <<<gT, t, 0, stream>>>(bufB, dis, bufA, total);
  k_edge_scatter<<<gS, t, 0, stream>>>(src, dst, dis, bufB, bufA, E);
  k_bias_relu   <<<gT, t, 0, stream>>>(bufA, b2, (float*)d_out, total);
}